// DMV_79190607004378
// MI455X (gfx1250) — compile-verified
//
#include <hip/hip_runtime.h>
#include <stdint.h>

#define NEGV  (-1000000000.0f)
#define INVLN2 1.4426950408889634f
#define LN2    0.6931471805599453f

typedef __attribute__((ext_vector_type(2))) float v2f;
typedef __attribute__((ext_vector_type(8))) float v8f;

__device__ __forceinline__ float fexp(float x) {
  return __builtin_amdgcn_exp2f(x * INVLN2);     // v_exp_f32
}
__device__ __forceinline__ float flog(float x) {
  return __builtin_amdgcn_logf(x) * LN2;         // v_log_f32
}

#if defined(__gfx1250__) && __has_builtin(__builtin_amdgcn_global_load_async_to_lds_b32)
#define USE_ASYNC_LDS 1
#endif
#if defined(__gfx1250__) && __has_builtin(__builtin_amdgcn_wmma_f32_16x16x4_f32)
#define USE_WMMA 1
#endif

// Wave-tile logsumexp over k=0..w-1 for 16 cells (cell = lane&15, lanes l and
// l+16 co-own a cell). Exact per-cell max (pass 1), then sum of exp(term-m)
// via chained V_WMMA_F32_16X16X4_F32 row-sums against an all-ones B (pass 2).
// term(k) is fully branchless: address-selects + unconditional ds_loads;
// out-of-range k produce garbage that is cndmask'ed to 0 before accumulation.
template <typename TermF>
__device__ __forceinline__ void wave_lse(int lane, int w, TermF term,
                                         float& pm_out, float& s_out) {
  const int g = lane >> 4;                 // 0: K=c,c+1   1: K=c+2,c+3
  float pm = -3.0e38f;
  for (int k = g; k < w; k += 2) pm = fmaxf(pm, term(k));
  pm = fmaxf(pm, __shfl_xor(pm, 16));      // exact per-cell max on both lanes
#ifdef USE_WMMA
  v8f acc = {};
  const v2f ones = {1.0f, 1.0f};
  for (int c = 0; c < w; c += 4) {
    const int k0 = c + 2 * g, k1 = k0 + 1;
    const float e0 = fexp(term(k0) - pm);  // unconditional: in-bounds/OOB->0
    const float e1 = fexp(term(k1) - pm);
    v2f a;
    a[0] = (k0 < w) ? e0 : 0.0f;           // v_cndmask, no branch
    a[1] = (k1 < w) ? e1 : 0.0f;
    // D[m][n] = sum_K A[m][K]*1 + C[m][n]  -> per-cell running sums
    acc = __builtin_amdgcn_wmma_f32_16x16x4_f32(false, a, false, ones,
                                                (short)0, acc, false, false);
  }
  // D layout: lanes 0-15 hold rows 0-7 (comp 0..7), lanes 16-31 rows 8-15.
  // Lane l<16 wants row l: l<8 -> own comp l; l>=8 -> partner's comp l&7.
  float x01 = (lane & 1) ? acc[1] : acc[0];
  float x23 = (lane & 1) ? acc[3] : acc[2];
  float x45 = (lane & 1) ? acc[5] : acc[4];
  float x67 = (lane & 1) ? acc[7] : acc[6];
  float x03 = (lane & 2) ? x23 : x01;
  float x47 = (lane & 2) ? x67 : x45;
  float sel = (lane & 4) ? x47 : x03;
  float oth = __shfl_xor(sel, 16);
  s_out = (lane & 8) ? oth : sel;
#else
  float p = 0.0f;
  for (int c = 0; c < w; c += 4) {
    const int k0 = c + 2 * g, k1 = k0 + 1;
    const float e0 = fexp(term(k0) - pm);
    const float e1 = fexp(term(k1) - pm);
    p += ((k0 < w) ? e0 : 0.0f) + ((k1 < w) ? e1 : 0.0f);
  }
  s_out = p + __shfl_xor(p, 16);
#endif
  pm_out = pm;
}

// Packed-triangular DMV inside algorithm, one 128-thread workgroup per batch row.
//   sC[4096] : j>h -> CR[h,j], j<h -> CL[h,j], diag = 0 (shared, both are 0)
//   sI[4096] : j>h -> IR[h,j], j<h -> IL[h,j] (diag unused)
//   sF[4096] : j>h -> FR[h,j], j<h -> FL[h,j] (diags split out: FRd/FLd)
//   sT[4096] : j>h -> trans_r[h,j], j<h -> trans_l[h,j]
__global__ __launch_bounds__(128)
void dmv_inside_kernel(const int* __restrict__ tag_array,
                       const int* __restrict__ len_array,
                       const float* __restrict__ root_param,
                       const float* __restrict__ trans_param,
                       const float* __restrict__ dec_param,
                       float* __restrict__ out)
{
  extern __shared__ float lds[];
  float* sC  = lds;
  float* sI  = lds + 4096;
  float* sF  = lds + 8192;
  float* sT  = lds + 12288;
  float* FRd = lds + 16384;        // stR_nc seeded diag of FR
  float* FLd = FRd + 64;           // stL_nc seeded diag of FL
  float* gRn = FRd + 128;          // goR_nc
  float* gRh = FRd + 192;          // goR_hc
  float* gLn = FRd + 256;          // goL_nc
  float* gLh = FRd + 320;          // goL_hc
  float* sRh = FRd + 384;          // stR_hc
  float* sLh = FRd + 448;          // stL_hc
  float* ROOT= FRd + 512;
  float* SC  = FRd + 576;
  int*   TAG = (int*)(FRd + 640);

  const int t    = threadIdx.x;
  const int wave = t >> 5;
  const int lane = t & 31;
  const int b    = blockIdx.x;

  // ---- load tags ----
  if (t < 64) TAG[t] = tag_array[b * 64 + t];
  __syncthreads();

  // ---- per-position parameter vectors + diagonals ----
  if (t < 64) {
    const int th = TAG[t];
    ROOT[t] = root_param[th];
    const float* d = dec_param + th * 8;   // [T,2,2,2] : t*8 + dir*4 + val*2 + dec
    gLn[t] = d[0];   // LEFT,NC,GO
    FLd[t] = d[1];   // LEFT,NC,STOP  (stL_nc)
    gLh[t] = d[2];   // LEFT,HC,GO
    sLh[t] = d[3];   // LEFT,HC,STOP
    gRn[t] = d[4];   // RIGHT,NC,GO
    FRd[t] = d[5];   // RIGHT,NC,STOP (stR_nc)
    gRh[t] = d[6];   // RIGHT,HC,GO
    sRh[t] = d[7];   // RIGHT,HC,STOP
    sC[t * 65] = 0.0f;                      // CR/CL diagonal
  }

  // ---- gather combined transition table sT[i][j] = tp[tag_i, tag_j, j>i] ----
  for (int idx = t; idx < 4096; idx += 128) {
    const int i = idx >> 6, j = idx & 63;
    const int off = TAG[i] * 128 + TAG[j] * 2 + (j > i ? 1 : 0);
#ifdef USE_ASYNC_LDS
    typedef __attribute__((address_space(1))) int gint_t;
    typedef __attribute__((address_space(3))) int lint_t;
    gint_t* gp = (gint_t*)(uintptr_t)(trans_param + off);
    lint_t* lp = (lint_t*)(uint32_t)(uintptr_t)(&sT[idx]);
    __builtin_amdgcn_global_load_async_to_lds_b32(gp, lp, 0, 0);
#else
    sT[idx] = trans_param[off];
#endif
  }
#ifdef USE_ASYNC_LDS
#if __has_builtin(__builtin_amdgcn_s_wait_asynccnt)
  __builtin_amdgcn_s_wait_asynccnt(0);
#else
  asm volatile("s_wait_asynccnt 0" ::: "memory");
#endif
#endif
  __syncthreads();

  // ---- width-serial DP ----
  for (int w = 1; w < 64; ++w) {
    const int nt = (64 - w + 15) >> 4;       // 16-cell tiles per direction
    const int mm = lane & 15;

    // ======== Stage I: attach a new dependent (IR / IL) ========
    for (int task = wave; task < 2 * nt; task += 4) {
      const int H = (task >> 1) * 16;
      if ((task & 1) == 0) {                 // RIGHT: cell h, e = h+w
        const int h = H + mm;
        const bool valid = (h + w) < 64;
        const int e = valid ? (h + w) : 63;
        const float* Crow = sC + h * 64;
        const float* Frow = sF + e * 64;
        const float* fdp  = FLd + e;
        const float gnc = gRn[h], ghc = gRh[h];
        auto term = [&](int k) {
          float cv = Crow[h + k];            // Crow[h] is diag = 0
          const float* fp = (k == w - 1) ? fdp : (Frow + h + k + 1);
          float gg = k ? ghc : gnc;
          return cv + gg + *fp;              // address-select + one ds_load
        };
        float pm, s;
        wave_lse(lane, w, term, pm, s);
        if (valid && lane < 16)
          sI[h * 64 + (h + w)] = sT[h * 64 + (h + w)] + pm + flog(s);
      } else {                               // LEFT: cell m0, hl = m0+w
        const int m0 = H + mm;
        const bool valid = (m0 + w) < 64;
        const int hl = valid ? (m0 + w) : 63;
        const float* Crow = sC + hl * 64;
        const float* Frow = sF + m0 * 64;
        const float* fdp  = FRd + m0;
        const float gnc = gLn[hl], ghc = gLh[hl];
        auto term = [&](int k) {
          float cv = Crow[hl - k];           // Crow[hl] is diag = 0
          const float* fp = (k == w - 1) ? fdp : (Frow + hl - k - 1);
          float gg = k ? ghc : gnc;
          return cv + gg + *fp;
        };
        float pm, s;
        wave_lse(lane, w, term, pm, s);
        if (valid && lane < 16)
          sI[(m0 + w) * 64 + m0] = sT[(m0 + w) * 64 + m0] + pm + flog(s);
      }
    }
    __syncthreads();

    // ======== Stage C: complete constituents (CR/FR, CL/FL) ========
    for (int task = wave; task < 2 * nt; task += 4) {
      const int H = (task >> 1) * 16;
      if ((task & 1) == 0) {                 // RIGHT
        const int h = H + mm;
        const bool valid = (h + w) < 64;
        const int e = valid ? (h + w) : 63;
        const float* Irow = sI + h * 64;
        const float* fdp  = FRd + e;
        auto term = [&](int k) {
          float iv = Irow[h + k + 1];
          const float* fp = (k == w - 1) ? fdp : (sF + (h + k + 1) * 64 + e);
          return iv + *fp;
        };
        float pm, s;
        wave_lse(lane, w, term, pm, s);
        if (valid && lane < 16) {
          const float cr = pm + flog(s);
          sC[h * 64 + (h + w)] = cr;
          sF[h * 64 + (h + w)] = cr + sRh[h];
        }
      } else {                               // LEFT
        const int m0 = H + mm;
        const bool valid = (m0 + w) < 64;
        const int hl = valid ? (m0 + w) : 63;
        const float* Irow = sI + hl * 64;
        const float* fdp  = FLd + m0;
        auto term = [&](int k) {
          float iv = Irow[hl - k - 1];
          const float* fp = (k == w - 1) ? fdp : (sF + (hl - k - 1) * 64 + m0);
          return iv + *fp;
        };
        float pm, s;
        wave_lse(lane, w, term, pm, s);
        if (valid && lane < 16) {
          const float cl = pm + flog(s);
          sC[(m0 + w) * 64 + m0] = cl;
          sF[(m0 + w) * 64 + m0] = cl + sLh[hl];
        }
      }
    }
    __syncthreads();
  }

  // ---- root attachment + final logsumexp over positions ----
  if (t < 64) {
    const int i = t;
    const int len = len_array[b];
    const int last = len - 1;
    float sc = NEGV;
    if (i < len) {
      const float fl0 = (i == 0) ? FLd[0] : sF[i * 64 + 0];          // FL[i,0]
      const float fre = (i == last) ? FRd[i] : sF[i * 64 + last];    // FR[i,last]
      sc = ROOT[i] + fl0 + fre;
    }
    SC[i] = sc;
  }
  __syncthreads();
  if (t == 0) {
    float m = SC[0];
    for (int i = 1; i < 64; ++i) m = fmaxf(m, SC[i]);
    float s = 0.0f;
    for (int i = 0; i < 64; ++i) s += fexp(SC[i] - m);
    out[b] = m + flog(s);
  }
}

extern "C" void kernel_launch(void* const* d_in, const int* in_sizes, int n_in,
                              void* d_out, int out_size, void* d_ws, size_t ws_size,
                              hipStream_t stream) {
  (void)n_in; (void)d_ws; (void)ws_size; (void)out_size;
  // inputs: 0=id_array(int,unused) 1=tag_array(int B*64) 2=len_array(int B)
  //         3=root_param(f32 64) 4=trans_param(f32 64*64*2) 5=dec_param(f32 64*8)
  const int*   tag   = (const int*)d_in[1];
  const int*   len   = (const int*)d_in[2];
  const float* root  = (const float*)d_in[3];
  const float* trans = (const float*)d_in[4];
  const float* dec   = (const float*)d_in[5];
  float* out = (float*)d_out;
  const int B = in_sizes[2];                     // batch = len_array size
  const size_t shmem = (4 * 4096 + 11 * 64) * sizeof(float);   // 68,352 B
  dmv_inside_kernel<<<dim3(B), dim3(128), shmem, stream>>>(tag, len, root, trans, dec, out);
}